// BondLengthConstraintEnergy_54803782697322
// MI455X (gfx1250) — compile-verified
//
#include <hip/hip_runtime.h>
#include <hip/hip_bf16.h>
#include <stdint.h>

// Problem constants (match reference setup_inputs)
#define B_      32
#define NCH_    8
#define NRES_   8192
#define APR_    3
#define NATOMS_ (B_ * NCH_ * NRES_ * APR_)   // 6,291,456
#define NOUT_   (B_ * NCH_ * NRES_)          // 2,097,152 (n_alt == 1)

#define EPS_     1e-8f
#define RAD2DEG_ 57.295779513082323f
#define TWO_PI_  6.2831853071795865f

typedef unsigned int u32x4 __attribute__((ext_vector_type(4)));
typedef int          i32x4 __attribute__((ext_vector_type(4)));
typedef int          i32x8 __attribute__((ext_vector_type(8)));

// ---------------------------------------------------------------------------
// TDM: stage a 2D f32 tensor (width x height, row-major, contiguous) into LDS.
// D# layout per CDNA5 ISA 8.3/8.4 (group0 128b, group1 256b; groups 2/3 zero
// since the tensor is 2D). 6-arg builtin form (clang-23 / therock headers).
// ---------------------------------------------------------------------------
__device__ __forceinline__ void tdm_load_2d_f32_to_lds(const float* gptr,
                                                       uint32_t lds_byte_off,
                                                       int width, int height) {
    uint64_t ga = (uint64_t)(uintptr_t)gptr;
    u32x4 g0;
    g0[0] = 1u;                                   // count=1 (valid), user mode
    g0[1] = lds_byte_off;                         // lds_addr [63:32]
    g0[2] = (uint32_t)(ga & 0xFFFFFFFFu);         // global_addr low
    g0[3] = (uint32_t)((ga >> 32) & 0x1FFFFFFu)   // global_addr[56:32]
          | (2u << 30);                           // type=2 ("image")
    i32x8 g1;
    g1[0] = 0x00020000;                           // data_size=2 -> 4 bytes
    g1[1] = (width  & 0xFFFF) << 16;              // tensor_dim0 (low 16)
    g1[2] = (height & 0xFFFF) << 16;              // tensor_dim1 (low 16)
    g1[3] = (width  & 0xFFFF) << 16;              // tile_dim0
    g1[4] = (height & 0xFFFF);                    // tile_dim1 (tile_dim2 = 0)
    g1[5] = width;                                // tensor_dim0_stride (low 32)
    g1[6] = 0;
    g1[7] = 0;
    i32x4 z4 = {0, 0, 0, 0};
    i32x8 z8 = {0, 0, 0, 0, 0, 0, 0, 0};
    __builtin_amdgcn_tensor_load_to_lds(g0, g1, z4, z4, z8, 0);
}

// ---------------------------------------------------------------------------
// Kernel 1: table = -1 (faithful to reference semantics)
// ---------------------------------------------------------------------------
__global__ void init_table_kernel(int* __restrict__ table, int n) {
    int i = blockIdx.x * blockDim.x + threadIdx.x;
    if (i < n) table[i] = -1;
}

// ---------------------------------------------------------------------------
// Kernel 2: scatter atom index into table[b][ch][r][at]
// ---------------------------------------------------------------------------
__global__ void scatter_table_kernel(const int* __restrict__ desc,
                                     int* __restrict__ table, int n) {
    int i = blockIdx.x * blockDim.x + threadIdx.x;
    if (i >= n) return;
    const int* row = desc + (size_t)i * 5;
    int b  = row[0];
    int ch = row[1];
    int r  = row[2];
    int at = row[4];
    table[(((b * NCH_) + ch) * NRES_ + r) * APR_ + at] = i;
}

// ---------------------------------------------------------------------------
// Kernel 3: per (b,ch,r) energy. r == NRES-1 writes 0 (also inits output).
// ---------------------------------------------------------------------------
__global__ void __launch_bounds__(256)
bond_energy_kernel(const int*   __restrict__ desc,
                   const float* __restrict__ coords,
                   const float* __restrict__ weight,
                   const float* __restrict__ mean,
                   const float* __restrict__ stdv,
                   const int*   __restrict__ table,
                   float*       __restrict__ out) {
    __shared__ float s_mean[64];   // 20x3 used
    __shared__ float s_std[64];

    // One wave issues two TDM loads; TENSORcnt is per-wave, then block barrier.
    if (threadIdx.x < 32) {
        tdm_load_2d_f32_to_lds(mean, (uint32_t)(uintptr_t)&s_mean[0], APR_, 20);
        tdm_load_2d_f32_to_lds(stdv, (uint32_t)(uintptr_t)&s_std[0],  APR_, 20);
        __builtin_amdgcn_s_wait_tensorcnt(0);
    }
    __syncthreads();

    int tid = blockIdx.x * blockDim.x + threadIdx.x;   // flat (b,ch,r)
    int r   = tid & (NRES_ - 1);

    float result = 0.0f;
    if (r < NRES_ - 1) {
        int t0 = tid * APR_;            // table slot of residue r
        int t1 = t0 + APR_;             // residue r+1 (same chain: r < NRES-1)
        int c_idx   = table[t0 + 2];    // C  of residue r
        int cac_idx = table[t0 + 1];    // CA of residue r
        int n_idx   = table[t1 + 0];    // N  of residue r+1
        int can_idx = table[t1 + 1];    // CA of residue r+1
        bool valid  = ((c_idx | cac_idx | n_idx | can_idx) >= 0);

        int cs   = c_idx   > 0 ? c_idx   : 0;
        int cacs = cac_idx > 0 ? cac_idx : 0;
        int ns   = n_idx   > 0 ? n_idx   : 0;
        int cans = can_idx > 0 ? can_idx : 0;

        const float* pc   = coords + (size_t)cs   * 3;
        const float* pn   = coords + (size_t)ns   * 3;
        const float* pcac = coords + (size_t)cacs * 3;
        const float* pcan = coords + (size_t)cans * 3;
        float cx = pc[0],   cy = pc[1],   cz = pc[2];
        float nx = pn[0],   ny = pn[1],   nz = pn[2];
        float ax = pcac[0], ay = pcac[1], az = pcac[2];   // CA(r)
        float bx = pcan[0], by = pcan[1], bz = pcan[2];   // CA(r+1)

        // v_cn = N - C ; v_nca_n = CA(r+1) - N ; v_cac_c = C - CA(r)
        float v0x = nx - cx, v0y = ny - cy, v0z = nz - cz;
        float v1x = bx - nx, v1y = by - ny, v1z = bz - nz;
        float v2x = cx - ax, v2y = cy - ay, v2z = cz - az;

        float n0 = sqrtf(v0x * v0x + v0y * v0y + v0z * v0z);   // bond length
        float n1 = sqrtf(v1x * v1x + v1y * v1y + v1z * v1z);
        float n2 = sqrtf(v2x * v2x + v2y * v2y + v2z * v2z);

        bool m1 = (n0 > EPS_) && (n1 > EPS_);
        bool m2 = (n2 > EPS_) && (n0 > EPS_);

        float cos1 = (v0x * v1x + v0y * v1y + v0z * v1z) / (n0 * n1 + EPS_);
        float cos2 = -(v2x * v0x + v2y * v0y + v2z * v0z) / (n2 * n0 + EPS_);
        cos1 = fminf(fmaxf(cos1, -1.0f), 1.0f);
        cos2 = fminf(fmaxf(cos2, -1.0f), 1.0f);
        float ang1 = acosf(cos1) * RAD2DEG_;
        float ang2 = acosf(cos2) * RAD2DEG_;

        float geom[3] = { n0, ang1, ang2 };

        int seq = desc[(size_t)cs * 5 + 3];   // resname of C atom's residue

        float score = 0.0f;
        #pragma unroll
        for (int k = 0; k < 3; ++k) {
            float m   = s_mean[seq * 3 + k];
            float s   = s_std[seq * 3 + k];
            float var = s * s;
            float den = sqrtf(TWO_PI_ * var);
            float d   = geom[k] - m;
            float num = expf(-(d * d) / (2.0f * var));
            // -(log(clip(num/den, EPS)) + log(den))
            score += -(logf(fmaxf(num / den, EPS_)) + logf(den));
        }

        float w      = weight[0];
        float factor = 1.0f - tanhf(-w);
        result = (valid && m1 && m2) ? score * factor : 0.0f;
    }
    out[tid] = result;
}

// ---------------------------------------------------------------------------
// Launch
// ---------------------------------------------------------------------------
extern "C" void kernel_launch(void* const* d_in, const int* in_sizes, int n_in,
                              void* d_out, int out_size, void* d_ws, size_t ws_size,
                              hipStream_t stream) {
    const int*   desc   = (const int*)d_in[0];    // atom_description (N_ATOMS,5) i32
    const float* coords = (const float*)d_in[1];  // (N_ATOMS,3) f32
    // d_in[2] = alternatives (N_ATOMS,1) f32 — only defines n_alt==1, unused
    const float* weight = (const float*)d_in[3];  // (1,) f32
    const float* mean   = (const float*)d_in[4];  // (20,3) f32
    const float* stdv   = (const float*)d_in[5];  // (20,3) f32

    int* table = (int*)d_ws;                      // NATOMS_ int32 = 25.2 MB

    const int threads = 256;                      // 8 wave32 per block
    int blocksA = (NATOMS_ + threads - 1) / threads;
    init_table_kernel<<<blocksA, threads, 0, stream>>>(table, NATOMS_);
    scatter_table_kernel<<<blocksA, threads, 0, stream>>>(desc, table, NATOMS_);

    int blocksE = NOUT_ / threads;                // 8192, exact
    bond_energy_kernel<<<blocksE, threads, 0, stream>>>(
        desc, coords, weight, mean, stdv, table, (float*)d_out);
}